// GATModule_29231547417083
// MI455X (gfx1250) — compile-verified
//
#include <hip/hip_runtime.h>
#include <hip/hip_bf16.h>
#include <math.h>

typedef float v2f __attribute__((ext_vector_type(2)));
typedef float v8f __attribute__((ext_vector_type(8)));

#define GAT_NEG_SLOPE 0.2f

// ---------------------------------------------------------------------------
// WMMA GEMM (f32): H[n x C] = A[n x K] @ B[K x C],  C = COLT*16.
// One wave32 computes a full 16 x C strip: the A fragment (dominant memory
// stream) is loaded ONCE per K-step and reused across COLT back-to-back
// V_WMMA_F32_16X16X4_F32 issues against the cache-resident weight matrix.
// A-matrix 16x4 layout (ISA 7.12.2): lanes 0-15 M=0..15, VGPR0=K0/VGPR1=K1;
// lanes 16-31 same M, VGPR0=K2/VGPR1=K3.  B 4x16: same with N instead of M.
// C/D 16x16: VGPR r -> M=r (lanes 0-15) / M=r+8 (lanes 16-31), N=lane&15.
// ---------------------------------------------------------------------------
template <int COLT>
__global__ void gat_gemm_wmma_f32_wide(const float* __restrict__ A,
                                       const float* __restrict__ B,
                                       float* __restrict__ H,
                                       int n, int K) {
    const int C    = COLT * 16;
    const int lane = threadIdx.x & 31;
    const int wave = (blockIdx.x * blockDim.x + threadIdx.x) >> 5;
    const int m0   = wave << 4;                 // 16-row strip per wave
    if (m0 >= n) return;                        // wave-uniform: EXEC stays all-1 for WMMA

    const int rlo = lane & 15;
    const int hi  = lane >> 4;                  // 0 or 1
    int arow = m0 + rlo; if (arow >= n) arow = n - 1;   // tail clamp (n%16==0 here)

    v8f acc[COLT] = {};
    const float* aptr  = A + (size_t)arow * K + 2 * hi;
    const float* bbase = B + (size_t)(2 * hi) * C + rlo;

    for (int k = 0; k < K; k += 4) {
        v2f a;
        a.x = aptr[0];                          // K = k + 2*hi
        a.y = aptr[1];                          // K = k + 2*hi + 1
        aptr += 4;
        const float* bp = bbase + (size_t)k * C;
#pragma unroll
        for (int ct = 0; ct < COLT; ++ct) {
            v2f b;
            b.x = bp[ct * 16];                  // B[k+2*hi   ][ct*16+rlo]
            b.y = bp[ct * 16 + C];              // B[k+2*hi+1 ][ct*16+rlo]
            acc[ct] = __builtin_amdgcn_wmma_f32_16x16x4_f32(
                false, a, false, b, (short)0, acc[ct], false, false);
        }
    }
#pragma unroll
    for (int ct = 0; ct < COLT; ++ct) {
#pragma unroll
        for (int r = 0; r < 8; ++r) {
            int orow = m0 + r + 8 * hi;
            if (orow < n) H[(size_t)orow * C + ct * 16 + rlo] = acc[ct][r];
        }
    }
}

// ---------------------------------------------------------------------------
// Per-node attention halves: as[n,h] = dot(H[n,h,:], a_src[h,:]) etc.
// ---------------------------------------------------------------------------
__global__ void gat_node_alpha(const float* __restrict__ H,
                               const float* __restrict__ a_src,
                               const float* __restrict__ a_dst,
                               float* __restrict__ as_, float* __restrict__ ad_,
                               int n, int heads, int hc) {
    int t = blockIdx.x * blockDim.x + threadIdx.x;
    if (t >= n * heads) return;
    int h = t % heads, node = t / heads;
    const float* hp = H + (size_t)node * heads * hc + (size_t)h * hc;
    const float* s = a_src + h * hc;
    const float* d = a_dst + h * hc;
    float sa = 0.f, da = 0.f;
    for (int c = 0; c < hc; ++c) { float v = hp[c]; sa += v * s[c]; da += v * d[c]; }
    as_[t] = sa; ad_[t] = da;
}

// order-preserving float <-> uint mapping for atomicMax
__device__ __forceinline__ unsigned gat_enc(float f) {
    unsigned u = __float_as_uint(f);
    return (u & 0x80000000u) ? ~u : (u | 0x80000000u);
}
__device__ __forceinline__ float gat_dec(unsigned k) {
    unsigned u = (k & 0x80000000u) ? (k & 0x7FFFFFFFu) : ~k;
    return __uint_as_float(u);
}

__device__ __forceinline__ void gat_edge(const long long* ei, int E, int e,
                                         int& src, int& dst) {
    if (e < E) { src = (int)ei[e]; dst = (int)ei[(size_t)E + e]; }
    else       { src = dst = e - E; }          // self-loop
}

// pass 1: raw score (leaky relu) + segment max (atomic max on encoded uint)
__global__ void gat_edge_score_max(const long long* __restrict__ ei, int E, int n,
                                   int heads,
                                   const float* __restrict__ as_,
                                   const float* __restrict__ ad_,
                                   float* __restrict__ esc,
                                   unsigned* __restrict__ mkey) {
    int t = blockIdx.x * blockDim.x + threadIdx.x;
    int ET = E + n;
    if (t >= ET * heads) return;
    int h = t % heads, e = t / heads;
    int src, dst; gat_edge(ei, E, e, src, dst);
    float v = as_[src * heads + h] + ad_[dst * heads + h];
    v = v > 0.f ? v : GAT_NEG_SLOPE * v;
    esc[t] = v;
    atomicMax(&mkey[dst * heads + h], gat_enc(v));
}

// pass 2: ee = exp(e - m[dst]); denom[dst] += ee; esc overwritten with ee
__global__ void gat_edge_exp_sum(const long long* __restrict__ ei, int E, int n,
                                 int heads,
                                 float* __restrict__ esc,
                                 const unsigned* __restrict__ mkey,
                                 float* __restrict__ den) {
    int t = blockIdx.x * blockDim.x + threadIdx.x;
    int ET = E + n;
    if (t >= ET * heads) return;
    int h = t % heads, e = t / heads;
    int src, dst; gat_edge(ei, E, e, src, dst);
    float ee = expf(esc[t] - gat_dec(mkey[dst * heads + h]));
    esc[t] = ee;
    atomicAdd(&den[dst * heads + h], ee);
}

// pass 3: out[dst, c] += h[src, c] * ee / (den[dst] + 1e-16)
__global__ void gat_edge_aggregate(const long long* __restrict__ ei, int E, int n,
                                   int heads, int hc,
                                   const float* __restrict__ esc,
                                   const float* __restrict__ den,
                                   const float* __restrict__ H,
                                   float* __restrict__ out) {
    int C = heads * hc;
    int t = blockIdx.x * blockDim.x + threadIdx.x;
    int ET = E + n;
    if (t >= ET * C) return;
    int c = t % C, e = t / C;
    int h = c / hc;
    int src, dst; gat_edge(ei, E, e, src, dst);
    float alpha = esc[e * heads + h] / (den[dst * heads + h] + 1e-16f);
    atomicAdd(&out[(size_t)dst * C + c], H[(size_t)src * C + c] * alpha);
}

// X[i] = elu(X[i] + b[c])
__global__ void gat_elu_bias(float* __restrict__ X, const float* __restrict__ b,
                             int total, int C) {
    int t = blockIdx.x * blockDim.x + threadIdx.x;
    if (t >= total) return;
    float v = X[t] + b[t % C];
    X[t] = v > 0.f ? v : (expf(v) - 1.f);
}

// X[i] += b[c]
__global__ void gat_add_bias(float* __restrict__ X, const float* __restrict__ b,
                             int total, int C) {
    int t = blockIdx.x * blockDim.x + threadIdx.x;
    if (t >= total) return;
    X[t] += b[t % C];
}

static inline int gridFor(long long work, int block) {
    return (int)((work + block - 1) / block);
}

extern "C" void kernel_launch(void* const* d_in, const int* in_sizes, int n_in,
                              void* d_out, int out_size, void* d_ws, size_t ws_size,
                              hipStream_t stream) {
    const float*     x   = (const float*)d_in[0];
    const long long* ei  = (const long long*)d_in[1];   // int64 [2, E]
    const float*     W1  = (const float*)d_in[2];       // [128, 64]
    const float*     a1s = (const float*)d_in[3];       // [2, 32]
    const float*     a1d = (const float*)d_in[4];
    const float*     b1  = (const float*)d_in[5];       // [64]
    const float*     W2  = (const float*)d_in[6];       // [64, 32]
    const float*     a2s = (const float*)d_in[7];       // [1, 32]
    const float*     a2d = (const float*)d_in[8];
    const float*     b2  = (const float*)d_in[9];       // [32]

    const int IN = 128, HID = 32, HEADS = 2, OUT = 32;
    const int C1 = HEADS * HID;                  // 64
    const int n  = in_sizes[0] / IN;             // 100000
    const int E  = in_sizes[1] / 2;              // 1.6M
    const int ET = E + n;                        // + self loops

    // ---- workspace layout (floats) ----
    float*    H1   = (float*)d_ws;               // n*64
    float*    AGG1 = H1   + (size_t)n * C1;      // n*64  (becomes elu(agg+b1) in place)
    float*    H2   = AGG1 + (size_t)n * C1;      // n*32
    float*    AS1  = H2   + (size_t)n * OUT;     // n*2
    float*    AD1  = AS1  + (size_t)n * HEADS;   // n*2
    unsigned* MK1  = (unsigned*)(AD1 + (size_t)n * HEADS);   // n*2
    float*    DEN1 = (float*)(MK1 + (size_t)n * HEADS);      // n*2
    float*    AS2  = DEN1 + (size_t)n * HEADS;   // n
    float*    AD2  = AS2  + (size_t)n;           // n
    unsigned* MK2  = (unsigned*)(AD2 + (size_t)n);           // n
    float*    DEN2 = (float*)(MK2 + (size_t)n);  // n
    float*    EE   = DEN2 + (size_t)n;           // ET*2 (layer1) / ET (layer2)

    const int BLK = 256;
    const int rowTiles = (n + 15) / 16;          // one wave per 16-row strip

    // ---- zero init (memset nodes are graph-capture safe) ----
    hipMemsetAsync(AGG1, 0, (size_t)n * C1 * sizeof(float), stream);
    hipMemsetAsync(AS1,  0, (size_t)n * 12 * sizeof(float), stream); // AS1..DEN2
    hipMemsetAsync(d_out, 0, (size_t)out_size * sizeof(float), stream);

    // ================= layer 1 =================
    // H1 = x @ W1  (WMMA f32, 16x64 strip per wave, A loaded once per K-step)
    gat_gemm_wmma_f32_wide<4><<<gridFor(rowTiles, 4), 128, 0, stream>>>(x, W1, H1, n, IN);
    gat_node_alpha<<<gridFor((long long)n * HEADS, BLK), BLK, 0, stream>>>(
        H1, a1s, a1d, AS1, AD1, n, HEADS, HID);
    gat_edge_score_max<<<gridFor((long long)ET * HEADS, BLK), BLK, 0, stream>>>(
        ei, E, n, HEADS, AS1, AD1, EE, MK1);
    gat_edge_exp_sum<<<gridFor((long long)ET * HEADS, BLK), BLK, 0, stream>>>(
        ei, E, n, HEADS, EE, MK1, DEN1);
    gat_edge_aggregate<<<gridFor((long long)ET * C1, BLK), BLK, 0, stream>>>(
        ei, E, n, HEADS, HID, EE, DEN1, H1, AGG1);
    gat_elu_bias<<<gridFor((long long)n * C1, BLK), BLK, 0, stream>>>(
        AGG1, b1, n * C1, C1);

    // ================= layer 2 =================
    // H2 = elu(h1) @ W2  (WMMA f32, 16x32 strip per wave)
    gat_gemm_wmma_f32_wide<2><<<gridFor(rowTiles, 4), 128, 0, stream>>>(AGG1, W2, H2, n, C1);
    gat_node_alpha<<<gridFor((long long)n, BLK), BLK, 0, stream>>>(
        H2, a2s, a2d, AS2, AD2, n, 1, OUT);
    gat_edge_score_max<<<gridFor((long long)ET, BLK), BLK, 0, stream>>>(
        ei, E, n, 1, AS2, AD2, EE, MK2);
    gat_edge_exp_sum<<<gridFor((long long)ET, BLK), BLK, 0, stream>>>(
        ei, E, n, 1, EE, MK2, DEN2);
    gat_edge_aggregate<<<gridFor((long long)ET * OUT, BLK), BLK, 0, stream>>>(
        ei, E, n, 1, OUT, EE, DEN2, H2, (float*)d_out);
    gat_add_bias<<<gridFor((long long)n * OUT, BLK), BLK, 0, stream>>>(
        (float*)d_out, b2, n * OUT, OUT);
}